// LocalNonLocalMultiheadSelfAttention_90091234001279
// MI455X (gfx1250) — compile-verified
//
#include <hip/hip_runtime.h>
#include <hip/hip_bf16.h>

// MI455X / gfx1250 implementation of local+non-local windowed multi-head attention.
// fp32 end-to-end (bandwidth-bound problem; working set fits in 192MB L2).
// Matrix math uses V_WMMA_F32_16X16X4_F32 (wave32 WMMA, K=4 fits head_dim=8).

typedef __attribute__((ext_vector_type(2))) float v2f;
typedef __attribute__((ext_vector_type(8))) float v8f;

#define BATCH 2
#define CDIM  64
#define HW    160
#define PIX   (HW * HW)          // 25600
#define QKVC  (3 * CDIM)         // 192
#define NWIN  400                // (160/8)^2
#define NWX   20                 // windows per row
#define SCALE 0.35355339059327373f   // 8^-0.5

static __device__ __forceinline__ v8f wmma4(v2f a, v2f b, v8f c) {
    // D = A(16x4,f32) * B(4x16,f32) + C(16x16,f32)
    return __builtin_amdgcn_wmma_f32_16x16x4_f32(
        /*neg_a=*/false, a, /*neg_b=*/false, b,
        /*c_mod=*/(short)0, c, /*reuse_a=*/false, /*reuse_b=*/false);
}

// ---------------------------------------------------------------------------
// K1: qkv = W_qkv(192x64) @ X(64 x B*PIX). One wave -> 16(M) x 64(N) tile.
// ---------------------------------------------------------------------------
__global__ __launch_bounds__(32) void qkv_gemm(const float* __restrict__ x,
                                               const float* __restrict__ wqkv,
                                               float* __restrict__ qkv0) {
    const int lane = threadIdx.x;
    const int lm = lane & 15, half = lane >> 4;
    const int m0 = blockIdx.y * 16;                 // output channel tile
    const long n0 = (long)blockIdx.x * 64;          // flat pixel tile (never crosses batch: 64|25600)
    const int batch = (int)(n0 / PIX);
    const int pbase = (int)(n0 % PIX);
    const float* xb = x + (size_t)batch * CDIM * PIX;

    v8f acc0 = {}, acc1 = {}, acc2 = {}, acc3 = {};
    for (int k0 = 0; k0 < CDIM; k0 += 4) {
        const int ka = k0 + 2 * half;
        v2f a;
        a.x = wqkv[(m0 + lm) * CDIM + ka];
        a.y = wqkv[(m0 + lm) * CDIM + ka + 1];
        const float* r0 = xb + (size_t)ka * PIX;
        const float* r1 = xb + (size_t)(ka + 1) * PIX;
        v2f b;
        b.x = r0[pbase + 0 * 16 + lm]; b.y = r1[pbase + 0 * 16 + lm];
        acc0 = wmma4(a, b, acc0);
        b.x = r0[pbase + 1 * 16 + lm]; b.y = r1[pbase + 1 * 16 + lm];
        acc1 = wmma4(a, b, acc1);
        b.x = r0[pbase + 2 * 16 + lm]; b.y = r1[pbase + 2 * 16 + lm];
        acc2 = wmma4(a, b, acc2);
        b.x = r0[pbase + 3 * 16 + lm]; b.y = r1[pbase + 3 * 16 + lm];
        acc3 = wmma4(a, b, acc3);
    }
    float* outb = qkv0 + (size_t)batch * QKVC * PIX;
    v8f accs[4] = {acc0, acc1, acc2, acc3};
#pragma unroll
    for (int t = 0; t < 4; ++t) {
        const int p = pbase + t * 16 + lm;
#pragma unroll
        for (int r = 0; r < 8; ++r) {
            const int o = m0 + r + 8 * half;        // D: row = vgpr + 8*(lane>=16)
            outb[(size_t)o * PIX + p] = accs[t][r];
        }
    }
}

// ---------------------------------------------------------------------------
// K2: depthwise 3x3, zero pad, groups = 192 (bandwidth kernel).
// ---------------------------------------------------------------------------
__global__ void dwconv(const float* __restrict__ qkv0,
                       const float* __restrict__ wdw,
                       float* __restrict__ qkv1) {
    const long idx = (long)blockIdx.x * 256 + threadIdx.x;
    const long total = (long)BATCH * QKVC * PIX;
    if (idx >= total) return;
    const int p  = (int)(idx % PIX);
    const int ch = (int)((idx / PIX) % QKVC);
    const int b  = (int)(idx / ((long)PIX * QKVC));
    const int y = p / HW, xp = p % HW;
    const float* w  = wdw + ch * 9;
    const float* in = qkv0 + ((size_t)b * QKVC + ch) * PIX;
    float s = 0.f;
#pragma unroll
    for (int dy = -1; dy <= 1; ++dy) {
        const int yy = y + dy;
        if (yy < 0 || yy >= HW) continue;
#pragma unroll
        for (int dx = -1; dx <= 1; ++dx) {
            const int xx = xp + dx;
            if (xx < 0 || xx >= HW) continue;
            s += in[yy * HW + xx] * w[(dy + 1) * 3 + (dx + 1)];
        }
    }
    qkv1[idx] = s;
}

// ---------------------------------------------------------------------------
// K3: local branch. One wave per (b, window, head): 64x64 attention, d=8.
// LDS padded to odd strides -> conflict-free on 64 banks.
// ---------------------------------------------------------------------------
__global__ __launch_bounds__(32) void local_attn(const float* __restrict__ qkv1,
                                                 float* __restrict__ attn) {
    __shared__ float Qs[64][9], Ks[64][9], Vs[64][9];
    __shared__ float S[64][65];
    __shared__ float invS[64];
    const int bid = blockIdx.x;                 // b*1600 + n*4 + h
    const int h = bid & 3;
    const int n = (bid >> 2) % NWIN;
    const int b = bid / (NWIN * 4);
    const int lane = threadIdx.x, lm = lane & 15, half = lane >> 4;
    const int wh = n / NWX, ww = n % NWX;
    const int cq = h * 8;                       // local branch: channels 0..31
    const size_t base = (size_t)b * QKVC * PIX;

    // position-fastest mapping: consecutive lanes -> contiguous window pixels
    for (int idx = lane; idx < 512; idx += 32) {
        const int d = idx >> 6, pos = idx & 63;
        const int pix = (wh * 8 + (pos >> 3)) * HW + ww * 8 + (pos & 7);
        Qs[pos][d] = qkv1[base + (size_t)(cq + d) * PIX + pix] * SCALE;
        Ks[pos][d] = qkv1[base + (size_t)(64 + cq + d) * PIX + pix];
        Vs[pos][d] = qkv1[base + (size_t)(128 + cq + d) * PIX + pix];
    }
    __syncthreads();

    // sim = Q @ K^T  (4x4 tiles of 16x16, K-dim 8 = two WMMA steps)
    for (int mt = 0; mt < 4; ++mt)
        for (int nt = 0; nt < 4; ++nt) {
            v8f acc = {};
#pragma unroll
            for (int k0 = 0; k0 < 8; k0 += 4) {
                const int ka = k0 + 2 * half;
                v2f a, bv;
                a.x = Qs[mt * 16 + lm][ka];  a.y = Qs[mt * 16 + lm][ka + 1];
                bv.x = Ks[nt * 16 + lm][ka]; bv.y = Ks[nt * 16 + lm][ka + 1];
                acc = wmma4(a, bv, acc);
            }
#pragma unroll
            for (int r = 0; r < 8; ++r)
                S[mt * 16 + r + 8 * half][nt * 16 + lm] = acc[r];
        }
    __syncthreads();

    // softmax (2 rows per lane): keep exp unnormalized, fold 1/sum into AV store
    for (int rr = 0; rr < 2; ++rr) {
        const int row = lane + rr * 32;
        float mx = -1e30f;
        for (int j = 0; j < 64; ++j) mx = fmaxf(mx, S[row][j]);
        float sum = 0.f;
        for (int j = 0; j < 64; ++j) { const float e = expf(S[row][j] - mx); S[row][j] = e; sum += e; }
        invS[row] = 1.f / sum;
    }
    __syncthreads();

    // O = attn(64x64) @ V(64x8); N padded 8->16 with zeros
    for (int mt = 0; mt < 4; ++mt) {
        v8f acc = {};
        for (int k0 = 0; k0 < 64; k0 += 4) {
            const int ka = k0 + 2 * half;
            v2f a, bv;
            a.x = S[mt * 16 + lm][ka]; a.y = S[mt * 16 + lm][ka + 1];
            bv.x = (lm < 8) ? Vs[ka][lm] : 0.f;
            bv.y = (lm < 8) ? Vs[ka + 1][lm] : 0.f;
            acc = wmma4(a, bv, acc);
        }
        if (lm < 8) {
#pragma unroll
            for (int r = 0; r < 8; ++r) {
                const int row = mt * 16 + r + 8 * half;
                const int pix = (wh * 8 + (row >> 3)) * HW + ww * 8 + (row & 7);
                attn[(size_t)b * CDIM * PIX + (size_t)(cq + lm) * PIX + pix] = acc[r] * invS[row];
            }
        }
    }
}

// ---------------------------------------------------------------------------
// K4: non-local branch. One wave per (b, intra-pos mm, head, 16-row stripe):
// sequence over 400 windows, d=8. sim stripe 16x400 in LDS (stride 401).
// ---------------------------------------------------------------------------
__global__ __launch_bounds__(32) void nonlocal_attn(const float* __restrict__ qkv1,
                                                    float* __restrict__ attn) {
    __shared__ float Ks[NWIN][9], Vs[NWIN][9], Qs[16][9];
    __shared__ float S[16][NWIN + 1];
    __shared__ float invS[16];
    const int bid = blockIdx.x;                 // b*6400 + mm*100 + h*25 + rt
    const int rt = bid % 25;
    const int h  = (bid / 25) & 3;
    const int mm = (bid / 100) % 64;
    const int b  = bid / 6400;
    const int lane = threadIdx.x, lm = lane & 15, half = lane >> 4;
    const int r = mm >> 3, c = mm & 7;
    const int ch = 32 + h * 8;                  // non-local branch: channels 32..63
    const size_t base = (size_t)b * QKVC * PIX;

    // position-fastest mapping for coalescing
    for (int idx = lane; idx < NWIN * 8; idx += 32) {
        const int d = idx / NWIN, j = idx % NWIN;
        const int pix = ((j / NWX) * 8 + r) * HW + (j % NWX) * 8 + c;
        Ks[j][d] = qkv1[base + (size_t)(64 + ch + d) * PIX + pix];
        Vs[j][d] = qkv1[base + (size_t)(128 + ch + d) * PIX + pix];
    }
    for (int idx = lane; idx < 128; idx += 32) {
        const int d = idx >> 4, i = idx & 15;
        const int j = rt * 16 + i;
        const int pix = ((j / NWX) * 8 + r) * HW + (j % NWX) * 8 + c;
        Qs[i][d] = qkv1[base + (size_t)(ch + d) * PIX + pix] * SCALE;
    }
    __syncthreads();

    // sim stripe: 16 rows x 400 cols
    for (int jt = 0; jt < 25; ++jt) {
        v8f acc = {};
#pragma unroll
        for (int k0 = 0; k0 < 8; k0 += 4) {
            const int ka = k0 + 2 * half;
            v2f a, bv;
            a.x = Qs[lm][ka];            a.y = Qs[lm][ka + 1];
            bv.x = Ks[jt * 16 + lm][ka]; bv.y = Ks[jt * 16 + lm][ka + 1];
            acc = wmma4(a, bv, acc);
        }
#pragma unroll
        for (int rr = 0; rr < 8; ++rr)
            S[rr + 8 * half][jt * 16 + lm] = acc[rr];
    }
    __syncthreads();

    // softmax: 2 lanes per row (200 cols each), combine via shfl_xor(16);
    // keep exp unnormalized, fold 1/sum into AV store.
    {
        const int row = lm;
        const int cb = half * 200;
        float mx = -1e30f;
        for (int j = 0; j < 200; ++j) mx = fmaxf(mx, S[row][cb + j]);
        mx = fmaxf(mx, __shfl_xor(mx, 16));
        float sum = 0.f;
        for (int j = 0; j < 200; ++j) { const float e = expf(S[row][cb + j] - mx); S[row][cb + j] = e; sum += e; }
        sum += __shfl_xor(sum, 16);
        if (half == 0) invS[row] = 1.f / sum;
    }
    __syncthreads();

    // O = attn(16x400) @ V(400x8)
    v8f acc = {};
    for (int k0 = 0; k0 < NWIN; k0 += 4) {
        const int ka = k0 + 2 * half;
        v2f a, bv;
        a.x = S[lm][ka]; a.y = S[lm][ka + 1];
        bv.x = (lm < 8) ? Vs[ka][lm] : 0.f;
        bv.y = (lm < 8) ? Vs[ka + 1][lm] : 0.f;
        acc = wmma4(a, bv, acc);
    }
    if (lm < 8) {
#pragma unroll
        for (int rr = 0; rr < 8; ++rr) {
            const int i = rr + 8 * half;
            const int j = rt * 16 + i;               // output row = window index
            const int pix = ((j / NWX) * 8 + r) * HW + (j % NWX) * 8 + c;
            attn[(size_t)b * CDIM * PIX + (size_t)(ch + lm) * PIX + pix] = acc[rr] * invS[i];
        }
    }
}

// ---------------------------------------------------------------------------
// K5: out = W_proj(64x64) @ attn(64 x B*PIX). Same structure as K1.
// ---------------------------------------------------------------------------
__global__ __launch_bounds__(32) void proj_gemm(const float* __restrict__ attn,
                                                const float* __restrict__ wproj,
                                                float* __restrict__ out) {
    const int lane = threadIdx.x;
    const int lm = lane & 15, half = lane >> 4;
    const int m0 = blockIdx.y * 16;
    const long n0 = (long)blockIdx.x * 64;
    const int batch = (int)(n0 / PIX);
    const int pbase = (int)(n0 % PIX);
    const float* ab = attn + (size_t)batch * CDIM * PIX;

    v8f acc0 = {}, acc1 = {}, acc2 = {}, acc3 = {};
    for (int k0 = 0; k0 < CDIM; k0 += 4) {
        const int ka = k0 + 2 * half;
        v2f a;
        a.x = wproj[(m0 + lm) * CDIM + ka];
        a.y = wproj[(m0 + lm) * CDIM + ka + 1];
        const float* r0 = ab + (size_t)ka * PIX;
        const float* r1 = ab + (size_t)(ka + 1) * PIX;
        v2f b;
        b.x = r0[pbase + 0 * 16 + lm]; b.y = r1[pbase + 0 * 16 + lm];
        acc0 = wmma4(a, b, acc0);
        b.x = r0[pbase + 1 * 16 + lm]; b.y = r1[pbase + 1 * 16 + lm];
        acc1 = wmma4(a, b, acc1);
        b.x = r0[pbase + 2 * 16 + lm]; b.y = r1[pbase + 2 * 16 + lm];
        acc2 = wmma4(a, b, acc2);
        b.x = r0[pbase + 3 * 16 + lm]; b.y = r1[pbase + 3 * 16 + lm];
        acc3 = wmma4(a, b, acc3);
    }
    float* ob = out + (size_t)batch * CDIM * PIX;
    v8f accs[4] = {acc0, acc1, acc2, acc3};
#pragma unroll
    for (int t = 0; t < 4; ++t) {
        const int p = pbase + t * 16 + lm;
#pragma unroll
        for (int r = 0; r < 8; ++r) {
            const int o = m0 + r + 8 * half;
            ob[(size_t)o * PIX + p] = accs[t][r];
        }
    }
}

// ---------------------------------------------------------------------------
extern "C" void kernel_launch(void* const* d_in, const int* in_sizes, int n_in,
                              void* d_out, int out_size, void* d_ws, size_t ws_size,
                              hipStream_t stream) {
    const float* x     = (const float*)d_in[0];
    const float* wqkv  = (const float*)d_in[1];
    const float* wdw   = (const float*)d_in[2];
    const float* wproj = (const float*)d_in[3];
    float* out = (float*)d_out;

    float* ws   = (float*)d_ws;
    float* qkv0 = ws;                                        // [2,192,25600]
    float* qkv1 = qkv0 + (size_t)BATCH * QKVC * PIX;         // [2,192,25600]
    float* attn = qkv1 + (size_t)BATCH * QKVC * PIX;         // [2,64,25600]

    {   // K1: 1x1 conv QKV (WMMA GEMM)
        dim3 g((BATCH * PIX) / 64, QKVC / 16);
        qkv_gemm<<<g, 32, 0, stream>>>(x, wqkv, qkv0);
    }
    {   // K2: depthwise 3x3
        const long total = (long)BATCH * QKVC * PIX;
        dwconv<<<(int)((total + 255) / 256), 256, 0, stream>>>(qkv0, wdw, qkv1);
    }
    {   // K3: local windowed attention (channels 0..31)
        local_attn<<<BATCH * NWIN * 4, 32, 0, stream>>>(qkv1, attn);
    }
    {   // K4: non-local attention (channels 32..63)
        nonlocal_attn<<<BATCH * 64 * 4 * 25, 32, 0, stream>>>(qkv1, attn);
    }
    {   // K5: 1x1 proj (WMMA GEMM)
        dim3 g((BATCH * PIX) / 64, CDIM / 16);
        proj_gemm<<<g, 32, 0, stream>>>(attn, wproj, out);
    }
}